// PlaneEncoding_90220083020079
// MI455X (gfx1250) — compile-verified
//
#include <hip/hip_runtime.h>
#include <cstdint>
#include <cstddef>

// ---------------------------------------------------------------------------
// PlaneEncoding: mip-pyramid trilinear fetch.
//   Phase 1: build levels 1..7 of the mip pyramid in d_ws, channel-INTERLEAVED
//            (texel-major, 8 contiguous floats per texel) -> 44.7 MB, L2-resident.
//   Phase 2: 2M-point gather: per-point mip select, bilinear at l0 & l1, lerp.
// Level 0 stays channel-major in the input (only ~1.6% of points touch it).
// ---------------------------------------------------------------------------

#define RES   2048
#define LVL   8
#define NCH   8
#define OFF1  4194304   // texel offset of level 1 in the reference pool layout

typedef float v2f __attribute__((ext_vector_type(2)));
typedef float v4f __attribute__((ext_vector_type(4)));
typedef int   vi4 __attribute__((vector_size(16)));   // matches builtin's V4i param

// OFFSETS[l] = sum_{i<l} (2048>>i)^2 = (16777216 - (16777216 >> 2l)) / 3  (exact)
__device__ __forceinline__ int level_off(int l) {
  return (16777216 - (16777216 >> (2 * l))) / 3;
}

// ----- CDNA5 async global->LDS path (guarded; falls back to plain loads) ----
#if defined(__AMDGCN__) && __has_builtin(__builtin_amdgcn_global_load_async_to_lds_b128)
#define HAVE_ASYNC_LDS 1
#else
#define HAVE_ASYNC_LDS 0
#endif

#if HAVE_ASYNC_LDS
__device__ __forceinline__ void async_copy_b128(const float* g, float* l) {
  __builtin_amdgcn_global_load_async_to_lds_b128(
      (__attribute__((address_space(1))) vi4*)(g),
      (__attribute__((address_space(3))) vi4*)(l),
      0, 0);
}
__device__ __forceinline__ void wait_async0() {
#if __has_builtin(__builtin_amdgcn_s_wait_asynccnt)
  __builtin_amdgcn_s_wait_asynccnt(0);
#else
  asm volatile("s_wait_asynccnt 0" ::: "memory");
#endif
}
#endif

// ---------------------------------------------------------------------------
// Level 1: read channel-major mat (8,2048,2048), 2x2 mean, write interleaved.
// ---------------------------------------------------------------------------
__global__ void __launch_bounds__(256)
mip_level1(const float* __restrict__ mat, float* __restrict__ pyr) {
  int t = blockIdx.x * 256 + threadIdx.x;   // t < 1024*1024, grid sized exactly
  int x = t & 1023;
  int y = t >> 10;
  const float* p0 = mat + ((size_t)(y << 1) << 11) + (size_t)(x << 1);
  v4f lo, hi;
#pragma unroll
  for (int c = 0; c < NCH; ++c) {
    const float* pc = p0 + (size_t)c * ((size_t)RES * RES);
    v2f a = *(const v2f*)pc;
    v2f b = *(const v2f*)(pc + RES);
    float m = (a[0] + a[1] + b[0] + b[1]) * 0.25f;
    if (c < 4) lo[c] = m; else hi[c - 4] = m;
  }
  v4f* o = (v4f*)(pyr + ((size_t)t << 3));
  o[0] = lo;
  o[1] = hi;
}

// ---------------------------------------------------------------------------
// Levels 2..7: interleaved -> interleaved 2x2 mean. Taps staged through LDS
// via CDNA5 async-tensor path when the toolchain exposes it.
// ---------------------------------------------------------------------------
__global__ void __launch_bounds__(256)
mip_down(float* __restrict__ pyr, int res_out, int shift,
         int in_off, int out_off, int n) {
  int t = blockIdx.x * 256 + threadIdx.x;
  if (t >= n) return;
  int x = t & (res_out - 1);
  int y = t >> shift;
  int res_in = res_out << 1;
  size_t e00 = (size_t)in_off + ((size_t)(y << 1) * res_in + (size_t)(x << 1));
  const float* g0 = pyr + (e00 << 3);                 // taps (0,0),(1,0): 16 floats
  const float* g1 = pyr + ((e00 + res_in) << 3);      // taps (0,1),(1,1): 16 floats

  float tap[32];
#if HAVE_ASYNC_LDS
  __shared__ float stage[256 * 32];                   // 32 KB, private slot per lane
  float* s = &stage[threadIdx.x * 32];
#pragma unroll
  for (int k = 0; k < 4; ++k) async_copy_b128(g0 + 4 * k, s + 4 * k);
#pragma unroll
  for (int k = 0; k < 4; ++k) async_copy_b128(g1 + 4 * k, s + 16 + 4 * k);
  wait_async0();
#pragma unroll
  for (int k = 0; k < 32; ++k) tap[k] = s[k];
#else
  const v4f* q0 = (const v4f*)g0;
  const v4f* q1 = (const v4f*)g1;
#pragma unroll
  for (int k = 0; k < 4; ++k) { v4f v = q0[k]; tap[4*k+0]=v[0]; tap[4*k+1]=v[1]; tap[4*k+2]=v[2]; tap[4*k+3]=v[3]; }
#pragma unroll
  for (int k = 0; k < 4; ++k) { v4f v = q1[k]; tap[16+4*k+0]=v[0]; tap[16+4*k+1]=v[1]; tap[16+4*k+2]=v[2]; tap[16+4*k+3]=v[3]; }
#endif

  v4f lo, hi;
#pragma unroll
  for (int c = 0; c < NCH; ++c) {
    float m = (tap[c] + tap[8 + c] + tap[16 + c] + tap[24 + c]) * 0.25f;
    if (c < 4) lo[c] = m; else hi[c - 4] = m;
  }
  v4f* o = (v4f*)(pyr + (((size_t)out_off + (size_t)t) << 3));
  o[0] = lo;
  o[1] = hi;
}

// ---------------------------------------------------------------------------
// Gather: per-point mip select + two bilinear fetches + level lerp.
// ---------------------------------------------------------------------------
__device__ __forceinline__ void level_params(int l, float xx, float xy,
                                             int& base, int& resl,
                                             float& w00, float& w10,
                                             float& w01, float& w11) {
  resl = RES >> l;
  float rf = (float)resl - 1.0f;
  float px = xx * rf;
  float py = xy * rf;
  int ix = (int)floorf(px);
  int iy = (int)floorf(py);
  ix = min(max(ix, 0), resl - 2);
  iy = min(max(iy, 0), resl - 2);
  float wx = fminf(fmaxf(px - (float)ix, 0.0f), 1.0f);
  float wy = fminf(fmaxf(py - (float)iy, 0.0f), 1.0f);
  base = level_off(l) + iy * resl + ix;
  w00 = (1.0f - wx) * (1.0f - wy);
  w10 = wx * (1.0f - wy);
  w01 = (1.0f - wx) * wy;
  w11 = wx * wy;
}

__device__ __forceinline__ void accum_pyr(const float* __restrict__ pyr,
                                          int base, int resl,
                                          float w00, float w10, float w01, float w11,
                                          float scale, v4f& accLo, v4f& accHi) {
  const v4f* q0 = (const v4f*)(pyr + ((size_t)(base - OFF1) << 3));
  const v4f* q1 = (const v4f*)(pyr + ((size_t)(base - OFF1 + resl) << 3));
  v4f a0 = q0[0], a1 = q0[1], b0 = q0[2], b1 = q0[3];   // taps 00, 10
  v4f c0 = q1[0], c1 = q1[1], d0 = q1[2], d1 = q1[3];   // taps 01, 11
  accLo += scale * (a0 * w00 + b0 * w10 + c0 * w01 + d0 * w11);
  accHi += scale * (a1 * w00 + b1 * w10 + c1 * w01 + d1 * w11);
}

__device__ __forceinline__ void accum_mat(const float* __restrict__ mat, int base,
                                          float w00, float w10, float w01, float w11,
                                          float scale, v4f& accLo, v4f& accHi) {
  const float* m = mat + base;   // level_off(0) == 0, resl == RES
#pragma unroll
  for (int c = 0; c < NCH; ++c) {
    const float* mc = m + (size_t)c * ((size_t)RES * RES);
    float f = mc[0] * w00 + mc[1] * w10 + mc[RES] * w01 + mc[RES + 1] * w11;
    if (c < 4) accLo[c] += scale * f; else accHi[c - 4] += scale * f;
  }
}

__global__ void __launch_bounds__(256)
gather_trilerp(const float* __restrict__ xs, const float* __restrict__ rs,
               const float* __restrict__ mat, const float* __restrict__ pyr,
               float* __restrict__ out, int nB) {
  int b = blockIdx.x * 256 + threadIdx.x;
  if (b >= nB) return;

  // Streamed inputs: non-temporal so they don't evict the L2-resident pyramid.
  v2f xv  = __builtin_nontemporal_load((const v2f*)xs + b);
  float rv = __builtin_nontemporal_load(rs + b);

  float rlog = log2f(2.0f * rv * (float)RES);
  rlog = fminf(fmaxf(rlog, 0.0f), (float)(LVL - 1));
  int l0 = (int)floorf(rlog);
  l0 = min(max(l0, 0), LVL - 1);
  int l1 = min(l0 + 1, LVL - 1);
  float wl = rlog - (float)l0;

  int base0, res0, base1, res1;
  float a00, a10, a01, a11, b00, b10, b01, b11;
  level_params(l0, xv[0], xv[1], base0, res0, a00, a10, a01, a11);
  level_params(l1, xv[0], xv[1], base1, res1, b00, b10, b01, b11);

  // Pre-touch level-l1 tap rows (always in interleaved pyramid since l1 >= 1)
  // while the l0 blend executes.  Lowers to global_prefetch_b8 on gfx1250.
  const float* pf0 = pyr + ((size_t)(base1 - OFF1) << 3);
  const float* pf1 = pyr + ((size_t)(base1 - OFF1 + res1) << 3);
  __builtin_prefetch(pf0, 0, 3);
  __builtin_prefetch(pf1, 0, 3);

  v4f accLo = {0.0f, 0.0f, 0.0f, 0.0f};
  v4f accHi = {0.0f, 0.0f, 0.0f, 0.0f};
  float s0 = 1.0f - wl;
  if (l0 == 0) accum_mat(mat, base0, a00, a10, a01, a11, s0, accLo, accHi);
  else         accum_pyr(pyr, base0, res0, a00, a10, a01, a11, s0, accLo, accHi);
  accum_pyr(pyr, base1, res1, b00, b10, b01, b11, wl, accLo, accHi);

  // Streamed output: non-temporal stores (write-once, no reuse).
  v4f* o = (v4f*)(out + ((size_t)b << 3));
  __builtin_nontemporal_store(accLo, o);
  __builtin_nontemporal_store(accHi, o + 1);
}

// ---------------------------------------------------------------------------
extern "C" void kernel_launch(void* const* d_in, const int* in_sizes, int n_in,
                              void* d_out, int out_size, void* d_ws, size_t ws_size,
                              hipStream_t stream) {
  const float* xs  = (const float*)d_in[0];   // (B, 2) f32
  const float* rs  = (const float*)d_in[1];   // (B,)   f32
  const float* mat = (const float*)d_in[2];   // (8, 2048, 2048) f32
  float* out = (float*)d_out;                 // (B, 8) f32
  float* pyr = (float*)d_ws;                  // interleaved levels 1..7: 44.7 MB
  int nB = in_sizes[1];

  // Level 1: transpose-downsample from channel-major input.
  mip_level1<<<(1 << 20) / 256, 256, 0, stream>>>(mat, pyr);

  // Levels 2..7: chained 2x2 means (stream-ordered dependencies).
  static const int off[LVL] = {0, 4194304, 5242880, 5505024,
                               5570560, 5586944, 5591040, 5592064};
  for (int l = 2; l < LVL; ++l) {
    int res_out = RES >> l;
    int shift   = 11 - l;
    int n       = res_out * res_out;
    mip_down<<<(n + 255) / 256, 256, 0, stream>>>(pyr, res_out, shift,
                                                  off[l - 1] - OFF1,
                                                  off[l] - OFF1, n);
  }

  // Gather phase (the hot kernel).
  gather_trilerp<<<(nB + 255) / 256, 256, 0, stream>>>(xs, rs, mat, pyr, out, nB);
}